// BoxTeacher_23167053594662
// MI455X (gfx1250) — compile-verified
//
#include <hip/hip_runtime.h>
#include <hip/hip_bf16.h>
#include <stdint.h>

#define B_ 8
#define M_ 100
#define N_ 32
#define HW_ 65536
#define IOU_THR_ 0.5f
#define SCORE_THR_ 0.0f

typedef __attribute__((ext_vector_type(4))) unsigned int u32x4;
typedef __attribute__((ext_vector_type(8))) unsigned int u32x8;

// ---------------------------------------------------------------------------
// Kernel 1: per-image greedy matching. One wave32 per batch image; lane j
// owns gt column j (N == 32 == wave32). Serial over the 100 score-ordered
// preds (faithful to the reference lax.scan), parallel over gts.
// ---------------------------------------------------------------------------
__global__ __launch_bounds__(32) void match_kernel(
    const float* __restrict__ pred_boxes,   // [B,M,4] xyxy
    const float* __restrict__ gt_boxes,     // [B,N,4] xyxy
    const float* __restrict__ pred_scores,  // [B,M]
    const float* __restrict__ mask_score,   // [B,M]
    float* __restrict__ out_flags,          // [B,N]
    float* __restrict__ out_gtm_f,          // [B,N] (matched index as float)
    float* __restrict__ out_biou,           // [B,N]
    int*   __restrict__ ws_gtm)             // [B,N] scratch for gather kernel
{
    __shared__ float s_pb[M_ * 4];
    __shared__ float s_sc[M_];
    __shared__ int   s_ord[M_];

    const int b    = blockIdx.x;
    const int lane = threadIdx.x;

    for (int i = lane; i < M_ * 4; i += 32) s_pb[i] = pred_boxes[b * M_ * 4 + i];
    for (int i = lane; i < M_; i += 32)     s_sc[i] = pred_scores[b * M_ + i];
    __syncthreads();

    // Stable descending argsort via rank counting (ties -> lower index first,
    // matching jnp.argsort(-scores) which is stable).
    for (int i = lane; i < M_; i += 32) {
        const float si = s_sc[i];
        int r = 0;
        for (int j = 0; j < M_; ++j) {
            const float sj = s_sc[j];
            r += (sj > si) || (sj == si && j < i);
        }
        s_ord[r] = i;
    }

    // This lane's gt box (registers for the whole greedy loop).
    const float g0 = gt_boxes[(b * N_ + lane) * 4 + 0];
    const float g1 = gt_boxes[(b * N_ + lane) * 4 + 1];
    const float g2 = gt_boxes[(b * N_ + lane) * 4 + 2];
    const float g3 = gt_boxes[(b * N_ + lane) * 4 + 3];
    const float areaG = (g2 - g0) * (g3 - g1);
    __syncthreads();

    int   gtm  = -1;    // reference init: -1
    float biou = 0.0f;  // reference init: 0

    for (int step = 0; step < M_; ++step) {
        const int i = s_ord[step];
        const float p0 = s_pb[i * 4 + 0], p1 = s_pb[i * 4 + 1];
        const float p2 = s_pb[i * 4 + 2], p3 = s_pb[i * 4 + 3];
        const float areaP = (p2 - p0) * (p3 - p1);

        const float lx = fmaxf(p0, g0), ly = fmaxf(p1, g1);
        const float rx = fminf(p2, g2), ry = fminf(p3, g3);
        const float w  = fmaxf(rx - lx, 0.0f), h = fmaxf(ry - ly, 0.0f);
        const float inter = w * h;
        const float iou   = inter / (areaP + areaG - inter);

        // Faithful quirk: gt available iff gtm[j] <= 0 (gt matched to pred 0
        // stays re-matchable).
        const float cand = (gtm <= 0 && iou >= IOU_THR_) ? iou : -1.0f;

        // Wave-wide argmax with first-index tie-break (associative combine,
        // so the xor butterfly leaves all 32 lanes with the same answer).
        float v  = cand;
        int   mi = lane;
        #pragma unroll
        for (int off = 16; off >= 1; off >>= 1) {
            const float v2 = __shfl_xor(v, off, 32);
            const int   m2 = __shfl_xor(mi, off, 32);
            if (v2 > v || (v2 == v && m2 < mi)) { v = v2; mi = m2; }
        }

        const bool doit = (s_sc[i] >= SCORE_THR_) && (v >= IOU_THR_);
        if (doit && lane == mi) { gtm = i; biou = v; }
    }

    const int  bn      = b * N_ + lane;
    const bool matched = gtm > -1;
    const int  idx     = matched ? gtm : 0;  // jnp.clip(gtm, 0)
    out_flags[bn] = matched ? mask_score[b * M_ + idx] : 0.0f;
    out_gtm_f[bn] = (float)gtm;
    out_biou[bn]  = biou;
    ws_gtm[bn]    = gtm;
}

// ---------------------------------------------------------------------------
// Kernel 2: mask gather. One block per (b, n). Matched tiles (256 KB) are
// moved by the Tensor Data Mover as 8 x 32 KB chunks pipelined over 4 LDS
// buffers. TDM completions are in-order per wave (loads and stores on one
// TENSORcnt stream), so s_wait_tensorcnt 3 before each issue guarantees the
// op issued 4-back is complete -- exactly the load->store data dependency
// and the buffer-reuse distance in the schedule below; up to 4 DMA transfers
// stay in flight per wave. Only wave 0 reaches the TDM instructions (EXEC is
// ignored for tensor ops, so other waves must branch around them).
// Unmatched tiles are zero-filled with vectorized stores by all 256 threads.
// ---------------------------------------------------------------------------
__global__ __launch_bounds__(256) void gather_kernel(
    const float* __restrict__ pred_masks,  // [B,M,HW]
    const int*   __restrict__ ws_gtm,      // [B,N]
    float* __restrict__ out_masks)         // [B,N,HW]
{
    __shared__ __align__(16) unsigned char stage[131072];  // 4 x 32 KB buffers

    const int bn = blockIdx.x;
    const int b  = bn >> 5;            // N_ == 32
    const int t  = threadIdx.x;
    const int m  = ws_gtm[bn];         // block-uniform

    float* dst = out_masks + (size_t)bn * HW_;

    if (m > -1) {
        if (t < 32) {  // wave 0 only: TDM issues regardless of EXEC
            const float* src = pred_masks + ((size_t)(b * M_ + m)) * HW_;

            // Uniform (SGPR) addresses for the descriptor.
            const uint64_t ps  = (uint64_t)(uintptr_t)src;
            const uint32_t plo = __builtin_amdgcn_readfirstlane((uint32_t)ps);
            const uint32_t phi = __builtin_amdgcn_readfirstlane((uint32_t)(ps >> 32));
            const uint64_t sga = ((uint64_t)phi << 32) | plo;

            const uint64_t pd  = (uint64_t)(uintptr_t)dst;
            const uint32_t dlo = __builtin_amdgcn_readfirstlane((uint32_t)pd);
            const uint32_t dhi = __builtin_amdgcn_readfirstlane((uint32_t)(pd >> 32));
            const uint64_t dga = ((uint64_t)dhi << 32) | dlo;

            // Low 32 bits of generic shared pointer == LDS byte offset.
            const uint32_t lds0 =
                __builtin_amdgcn_readfirstlane((uint32_t)(uintptr_t)&stage[0]);

            // D# group 1 (256 b), shared by all 8 transfers:
            //   data_size=3 (8 B units)           -> w0[17:16]
            //   tensor_dim0=4096 (32 KB chunk)    -> bits[79:48]  (w1[31:16])
            //   tensor_dim1=1                     -> bits[111:80] (w2[31:16])
            //   tile_dim0=4096                    -> bits[127:112](w3[31:16])
            //   tile_dim1=tile_dim2=0 (1-D tile)  -> w4
            //   tensor_dim0_stride=4096           -> bits[207:160](w5)
            const u32x8 g1 = { 3u << 16,      // data_size = 8 B
                               0x1000u << 16, // tensor_dim0 lo16
                               1u << 16,      // tensor_dim1 lo16
                               0x1000u << 16, // tile_dim0
                               0u, 4096u, 0u, 0u };

            #define TDM_LOAD(c)                                                    \
                do {                                                               \
                    const uint64_t sg_ = sga + (uint64_t)(c) * 32768u;             \
                    const u32x4 g0_ = { 1u, lds0 + ((c) & 3u) * 32768u,            \
                                        (uint32_t)sg_,                             \
                                        ((uint32_t)(sg_ >> 32) & 0x1FFFFFFu) |     \
                                            0x80000000u };                         \
                    asm volatile("tensor_load_to_lds %0, %1"                       \
                                 :: "s"(g0_), "s"(g1) : "memory");                 \
                } while (0)

            #define TDM_STORE(c)                                                   \
                do {                                                               \
                    const uint64_t dg_ = dga + (uint64_t)(c) * 32768u;             \
                    const u32x4 g0_ = { 1u, lds0 + ((c) & 3u) * 32768u,            \
                                        (uint32_t)dg_,                             \
                                        ((uint32_t)(dg_ >> 32) & 0x1FFFFFFu) |     \
                                            0x80000000u };                         \
                    asm volatile("tensor_store_from_lds %0, %1"                    \
                                 :: "s"(g0_), "s"(g1) : "memory");                 \
                } while (0)

            // Pipelined schedule, reuse distance 4, in-order completions:
            //   L0 L1 L2 L3 | w3 S0 .. w3 S3 | w3 L4 .. w3 L7 | w3 S4 .. S7 | w0
            #pragma unroll
            for (int c = 0; c < 4; ++c) TDM_LOAD(c);
            #pragma unroll
            for (int c = 0; c < 4; ++c) {
                __builtin_amdgcn_s_wait_tensorcnt(3);  // L(c) complete
                TDM_STORE(c);
            }
            #pragma unroll
            for (int c = 4; c < 8; ++c) {
                __builtin_amdgcn_s_wait_tensorcnt(3);  // S(c-4) complete
                TDM_LOAD(c);
            }
            #pragma unroll
            for (int c = 4; c < 8; ++c) {
                __builtin_amdgcn_s_wait_tensorcnt(3);  // L(c) complete
                TDM_STORE(c);
            }
            __builtin_amdgcn_s_wait_tensorcnt(0);      // drain

            #undef TDM_LOAD
            #undef TDM_STORE
        }
    } else {
        // Unmatched gt: zero-fill the tile (the common case for random boxes).
        const float4 z = make_float4(0.f, 0.f, 0.f, 0.f);
        float4* d4 = (float4*)dst;
        #pragma unroll 4
        for (int i = t; i < HW_ / 4; i += 256) d4[i] = z;
    }
}

// ---------------------------------------------------------------------------
// Launch. Output layout (flat, return order):
//   [gt_masks B*N*HW][flags B*N][gtm B*N (as float)][biou B*N]
// ---------------------------------------------------------------------------
extern "C" void kernel_launch(void* const* d_in, const int* in_sizes, int n_in,
                              void* d_out, int out_size, void* d_ws, size_t ws_size,
                              hipStream_t stream) {
    const float* pred_boxes  = (const float*)d_in[0];
    const float* gt_boxes    = (const float*)d_in[1];
    const float* pred_scores = (const float*)d_in[2];
    const float* pred_masks  = (const float*)d_in[3];
    const float* mask_score  = (const float*)d_in[4];

    float* out       = (float*)d_out;
    float* out_masks = out;                                   // B*N*HW
    float* out_flags = out + (size_t)B_ * N_ * HW_;           // B*N
    float* out_gtm   = out_flags + B_ * N_;                   // B*N
    float* out_biou  = out_gtm + B_ * N_;                     // B*N
    int*   ws_gtm    = (int*)d_ws;                            // B*N ints

    match_kernel<<<B_, 32, 0, stream>>>(pred_boxes, gt_boxes, pred_scores,
                                        mask_score, out_flags, out_gtm,
                                        out_biou, ws_gtm);
    gather_kernel<<<B_ * N_, 256, 0, stream>>>(pred_masks, ws_gtm, out_masks);
}